// AttentionModel_58171037057782
// MI455X (gfx1250) — compile-verified
//
#include <hip/hip_runtime.h>
#include <math.h>

// Problem sizes (fixed by the reference).
#define S_DIM 2048
#define B_DIM 32
#define H_DIM 1024

typedef __attribute__((ext_vector_type(2))) float v2f;
typedef __attribute__((ext_vector_type(8))) float v8f;

// ---------------------------------------------------------------------------
// Kernel 1: u[h] = sum_k v[k] * We[k][h] = sum_k v[k] * attn_W[k][H + h]
// 4 MB read, coalesced across h (adjacent threads -> adjacent columns).
// ---------------------------------------------------------------------------
__global__ __launch_bounds__(256) void compute_u_kernel(
    const float* __restrict__ attn_W,  // (H, 2H)
    const float* __restrict__ v,       // (1, H)
    float* __restrict__ u)             // (H,)
{
    const int h = blockIdx.x * 256 + threadIdx.x;  // 0..H-1
    float acc = 0.0f;
    const float* col = attn_W + H_DIM + h;  // We column h, row stride 2H
#pragma unroll 4
    for (int k = 0; k < H_DIM; ++k) {
        acc = fmaf(v[k], col[(size_t)k * (2 * H_DIM)], acc);
    }
    u[h] = acc;
}

// ---------------------------------------------------------------------------
// Kernel 2: energy[b][s] = sum_h enc[s][b][h] * u[h]   via V_WMMA_F32_16X16X4_F32
//
// One wave per (b, 16-wide s tile).  Per 4-wide h chunk:
//   A (16x4, MxK): A[m,k] = u[h0+k]  (broadcast over rows m)
//       lane L, vgpr j holds A[M=L%16, K = 2*(L/16)+j]  ->  u[h0 + koff + j]
//   B (4x16, KxN): B[k,n] = enc[s0+n][b][h0+k]
//       lane L, vgpr j holds B[K = 2*(L/16)+j, N=L%16] ->  enc_row(L%16)[h0+koff+j]
//   D (16x16): every row m equals energy[s0+n]; row 0 lives in acc[0], lanes 0..15.
// Four independent accumulators (h0, h0+4, h0+8, h0+12) hide the WMMA latency;
// the elementwise sum of their D matrices is the full reduction.
// ---------------------------------------------------------------------------
__global__ __launch_bounds__(256) void energy_wmma_kernel(
    const float* __restrict__ enc,  // (S, B, H)
    const float* __restrict__ u,    // (H,)
    float* __restrict__ energy)     // (B, S)
{
    __shared__ float su[H_DIM];
    // Cooperative fill of u into LDS (4 KB).
#pragma unroll
    for (int i = 0; i < H_DIM / 256; ++i)
        su[threadIdx.x + i * 256] = u[threadIdx.x + i * 256];
    __syncthreads();

    const int lane = threadIdx.x & 31;
    const int wave = threadIdx.x >> 5;                 // 8 waves per block
    const int task = blockIdx.x * 8 + wave;            // 0 .. B*(S/16)-1
    const int b    = task >> 7;                        // S/16 = 128 tiles per b
    const int s0   = (task & 127) << 4;

    const int koff = (lane >> 4) << 1;                 // 0 for lanes 0-15, 2 for 16-31
    const int row  = s0 + (lane & 15);                 // s index this lane serves
    const float* rp = enc + ((size_t)row * B_DIM + b) * H_DIM;

    v8f c0 = {}, c1 = {}, c2 = {}, c3 = {};
    for (int h = 0; h < H_DIM; h += 16) {
        // u chunks from LDS (ds_load_b64), enc chunks from global (global_load_b64).
        v2f a0 = *(const v2f*)(&su[h + 0  + koff]);
        v2f a1 = *(const v2f*)(&su[h + 4  + koff]);
        v2f a2 = *(const v2f*)(&su[h + 8  + koff]);
        v2f a3 = *(const v2f*)(&su[h + 12 + koff]);
        v2f b0 = *(const v2f*)(rp + h + 0  + koff);
        v2f b1 = *(const v2f*)(rp + h + 4  + koff);
        v2f b2 = *(const v2f*)(rp + h + 8  + koff);
        v2f b3 = *(const v2f*)(rp + h + 12 + koff);
        // 8 args: (neg_a, A, neg_b, B, c_mod, C, reuse_a, reuse_b)
        c0 = __builtin_amdgcn_wmma_f32_16x16x4_f32(false, a0, false, b0, (short)0, c0, false, false);
        c1 = __builtin_amdgcn_wmma_f32_16x16x4_f32(false, a1, false, b1, (short)0, c1, false, false);
        c2 = __builtin_amdgcn_wmma_f32_16x16x4_f32(false, a2, false, b2, (short)0, c2, false, false);
        c3 = __builtin_amdgcn_wmma_f32_16x16x4_f32(false, a3, false, b3, (short)0, c3, false, false);
    }
    // D row 0 (M=0) = VGPR 0, lanes 0..15 hold N=0..15.
    const float e = c0[0] + c1[0] + c2[0] + c3[0];
    if (lane < 16) {
        energy[(size_t)b * S_DIM + s0 + lane] = e;
    }
}

// ---------------------------------------------------------------------------
// Kernel 3: out[b][0][s] = softmax over s of energy[b][s].  One block per b.
// ---------------------------------------------------------------------------
__global__ __launch_bounds__(256) void softmax_kernel(
    const float* __restrict__ energy,  // (B, S)
    float* __restrict__ out)           // (B, 1, S)
{
    const int b = blockIdx.x;
    const float* e = energy + (size_t)b * S_DIM;
    __shared__ float sred[8];
    const int wid  = threadIdx.x >> 5;
    const int lane = threadIdx.x & 31;

    // ---- max reduction ----
    float m = -3.402823466e38f;
    for (int s = threadIdx.x; s < S_DIM; s += 256) m = fmaxf(m, e[s]);
    for (int off = 16; off > 0; off >>= 1) m = fmaxf(m, __shfl_xor(m, off, 32));
    if (lane == 0) sred[wid] = m;
    __syncthreads();
    if (threadIdx.x == 0) {
        float t = sred[0];
        for (int i = 1; i < 8; ++i) t = fmaxf(t, sred[i]);
        sred[0] = t;
    }
    __syncthreads();
    m = sred[0];
    __syncthreads();

    // ---- sum of exp ----
    float sum = 0.0f;
    for (int s = threadIdx.x; s < S_DIM; s += 256) sum += expf(e[s] - m);
    for (int off = 16; off > 0; off >>= 1) sum += __shfl_xor(sum, off, 32);
    if (lane == 0) sred[wid] = sum;
    __syncthreads();
    if (threadIdx.x == 0) {
        float t = 0.0f;
        for (int i = 0; i < 8; ++i) t += sred[i];
        sred[0] = t;
    }
    __syncthreads();
    const float inv = 1.0f / sred[0];

    // ---- normalize ----
    for (int s = threadIdx.x; s < S_DIM; s += 256)
        out[(size_t)b * S_DIM + s] = expf(e[s] - m) * inv;
}

// ---------------------------------------------------------------------------
// Launch.  Inputs (setup_inputs order):
//   d_in[0] encoder_outputs (S,B,H) f32
//   d_in[1] rnn_hidden      (1,B,H) f32   -- unused (softmax shift invariance)
//   d_in[2] attn_W          (H,2H)  f32
//   d_in[3] attn_b          (H,)    f32   -- unused (softmax shift invariance)
//   d_in[4] v               (1,H)   f32
// Workspace: u (H floats) at ws+0, energy (B*S floats) at ws+4096.
// ---------------------------------------------------------------------------
extern "C" void kernel_launch(void* const* d_in, const int* in_sizes, int n_in,
                              void* d_out, int out_size, void* d_ws, size_t ws_size,
                              hipStream_t stream) {
    (void)in_sizes; (void)n_in; (void)out_size; (void)ws_size;
    const float* enc = (const float*)d_in[0];
    const float* W   = (const float*)d_in[2];
    const float* v   = (const float*)d_in[4];
    float* out = (float*)d_out;

    float* u      = (float*)d_ws;                          // H floats
    float* energy = (float*)((char*)d_ws + 4096);          // B*S floats

    // 1) u = v @ We
    compute_u_kernel<<<H_DIM / 256, 256, 0, stream>>>(W, v, u);

    // 2) energies via f32 WMMA: B*(S/16) waves, 8 waves (256 threads) per block
    const int n_tasks = B_DIM * (S_DIM / 16);              // 4096
    energy_wmma_kernel<<<n_tasks / 8, 256, 0, stream>>>(enc, u, energy);

    // 3) softmax per batch row
    softmax_kernel<<<B_DIM, 256, 0, stream>>>(energy, out);
}